// SolventLayer_26706106647095
// MI455X (gfx1250) — compile-verified
//
#include <hip/hip_runtime.h>
#include <hip/hip_bf16.h>

// ---------------------------------------------------------------------------
// SolventLayer forward for MI455X (gfx1250).
// Dense math on V_WMMA_F32_16X16X4_F32 (fp32 matrix pipe, wave32).
// Weight staging uses GLOBAL_LOAD_ASYNC_TO_LDS_B128 (ASYNCcnt) — builtin is
// present in ROCm 7.2 with int4 address-space-typed pointer params.
// Working set (~150MB) is L2-resident (192MB). Deterministic fixed-order
// reductions (no FP atomics).
// ---------------------------------------------------------------------------

typedef __attribute__((ext_vector_type(2))) float v2f;
typedef __attribute__((ext_vector_type(4))) float v4f;
typedef __attribute__((ext_vector_type(8))) float v8f;
typedef __attribute__((ext_vector_type(4))) int v4i;

#define HID 64
#define NGRAPH 4096
#define NODE_DIM 74

#if defined(__HIP_DEVICE_COMPILE__) && \
    __has_builtin(__builtin_amdgcn_global_load_async_to_lds_b128)
#define HAVE_ASYNC_LDS 1
typedef __attribute__((address_space(1))) v4i g_v4i;  // global (AS1) int4
typedef __attribute__((address_space(3))) v4i l_v4i;  // LDS (AS3) int4
#else
#define HAVE_ASYNC_LDS 0
#endif

// Copy 16 bytes global -> LDS (async direct-to-LDS path when available).
static __device__ __forceinline__ void stage_b128(float* lds_dst,
                                                  const float* gsrc) {
#if HAVE_ASYNC_LDS
  __builtin_amdgcn_global_load_async_to_lds_b128(
      (g_v4i*)gsrc, (l_v4i*)lds_dst, /*offset=*/0, /*cpol=*/0);
#else
  *(v4f*)lds_dst = *(const v4f*)gsrc;
#endif
}

// Make this wave's outstanding async LDS copies complete (caller barriers).
static __device__ __forceinline__ void stage_commit() {
#if HAVE_ASYNC_LDS
#if __has_builtin(__builtin_amdgcn_s_wait_asynccnt)
  __builtin_amdgcn_s_wait_asynccnt(0);
#else
  asm volatile("s_wait_asynccnt 0" ::: "memory");
#endif
#endif
}

static __device__ __forceinline__ v8f wmma_f32_k4(v2f a, v2f b, v8f c) {
  // D = A(16x4,f32) * B(4x16,f32) + C(16x16,f32)
  return __builtin_amdgcn_wmma_f32_16x16x4_f32(
      /*neg_a=*/false, a, /*neg_b=*/false, b,
      /*c_mod=*/(short)0, c, /*reuse_a=*/false, /*reuse_b=*/false);
}

static __device__ __forceinline__ v8f v8f_zero() {
  v8f z;
#pragma unroll
  for (int j = 0; j < 8; ++j) z[j] = 0.0f;
  return z;
}

// ---------------------------------------------------------------------------
// hs0 = solv_feats @ emb_W + emb_b        [Ns,74] @ [74,64]
// ---------------------------------------------------------------------------
__global__ __launch_bounds__(256) void embed_kernel(
    const float* __restrict__ feats, const float* __restrict__ W,
    const float* __restrict__ bias, float* __restrict__ out, int Ns) {
  __shared__ __align__(16) float Wl[NODE_DIM * HID];  // 4736 = 1184 x float4
  __shared__ __align__(16) float bl[HID];
  int t = threadIdx.x;
  for (int i = t; i < (NODE_DIM * HID) / 4; i += 256)
    stage_b128(&Wl[i * 4], &W[i * 4]);
  if (t < HID / 4) stage_b128(&bl[t * 4], &bias[t * 4]);
  stage_commit();
  __syncthreads();
  int c = t & 63;
  int nbase = blockIdx.x * 16 + (t >> 6);
  for (int g = 0; g < 4; ++g) {
    int n = nbase + g * 4;
    if (n < Ns) {
      const float* f = feats + (size_t)n * NODE_DIM;
      float acc = bl[c];
#pragma unroll
      for (int j = 0; j < NODE_DIM; ++j) acc += f[j] * Wl[j * HID + c];
      out[(size_t)n * HID + c] = acc;
    }
  }
}

// ---------------------------------------------------------------------------
// Ring aggregation: deg==3 for every node -> agg = (x + x_prev + x_next)/3.
// ---------------------------------------------------------------------------
__global__ __launch_bounds__(256) void agg_kernel(
    const float* __restrict__ hs, const int* __restrict__ seg,
    const int* __restrict__ pos, float* __restrict__ out, int Ns) {
  int idx = blockIdx.x * 256 + threadIdx.x;
  int n = idx >> 6, c = idx & 63;
  if (n >= Ns) return;
  int s = seg[n], p = pos[n];
  int cnt = 16 + (s % 49);
  int b0 = n - p;
  int pn = b0 + (p == 0 ? cnt - 1 : p - 1);
  int nx = b0 + (p == cnt - 1 ? 0 : p + 1);
  out[idx] =
      (hs[idx] + hs[(size_t)pn * HID + c] + hs[(size_t)nx * HID + c]) *
      (1.0f / 3.0f);
}

// ---------------------------------------------------------------------------
// Y = act(X @ W [+ bias]) for W 64x64; optional BN partial stats.
// 8 waves; wave = 16 rows x 64 cols as 4 WMMA tiles, K=64 in 16 k4 steps.
// ---------------------------------------------------------------------------
__global__ __launch_bounds__(256) void mm64_kernel(
    const float* __restrict__ X, const float* __restrict__ W,
    const float* __restrict__ bias, float* __restrict__ Y,
    float* __restrict__ statsPart, int rowsValid, int doRelu) {
  __shared__ __align__(16) float Wl[HID * HID];
  __shared__ float sbuf[8 * 4 * 32];
  __shared__ float qbuf[8 * 4 * 32];
  int t = threadIdx.x;
  for (int i = t; i < (HID * HID) / 4; i += 256)
    stage_b128(&Wl[i * 4], &W[i * 4]);
  stage_commit();
  __syncthreads();

  int w = t >> 5, lane = t & 31;
  int li = lane & 15, hi = lane >> 4, koff = hi * 2;
  int r0 = blockIdx.x * 128 + w * 16;
  int xr = r0 + li;
  if (xr >= rowsValid) xr = rowsValid - 1;  // never read past real input
  const float* Xr = X + (size_t)xr * HID;

  v8f acc[4];
#pragma unroll
  for (int nt = 0; nt < 4; ++nt) acc[nt] = v8f_zero();

#pragma unroll
  for (int k0 = 0; k0 < HID; k0 += 4) {
    v2f a;
    a.x = Xr[k0 + koff];
    a.y = Xr[k0 + koff + 1];
#pragma unroll
    for (int nt = 0; nt < 4; ++nt) {
      v2f b;
      b.x = Wl[(k0 + koff) * HID + nt * 16 + li];
      b.y = Wl[(k0 + koff + 1) * HID + nt * 16 + li];
      acc[nt] = wmma_f32_k4(a, b, acc[nt]);
    }
  }

#pragma unroll
  for (int nt = 0; nt < 4; ++nt) {
    int col = nt * 16 + li;
    float bv = bias ? bias[col] : 0.0f;
    float s = 0.0f, q = 0.0f;
#pragma unroll
    for (int j = 0; j < 8; ++j) {
      int row = r0 + j + 8 * hi;
      float v = acc[nt][j] + bv;
      if (doRelu) v = fmaxf(v, 0.0f);
      Y[(size_t)row * HID + col] = v;  // padded buffer: always in bounds
      if (row < rowsValid) { s += v; q += v * v; }
    }
    if (statsPart) {
      sbuf[(w * 4 + nt) * 32 + lane] = s;
      qbuf[(w * 4 + nt) * 32 + lane] = q;
    }
  }
  if (statsPart) {
    __syncthreads();
    if (t < 64) {
      int nt = t >> 4, lc = t & 15;
      float S = 0.0f, Q = 0.0f;
      for (int ww = 0; ww < 8; ++ww) {  // fixed order -> deterministic
        S += sbuf[(ww * 4 + nt) * 32 + lc] + sbuf[(ww * 4 + nt) * 32 + lc + 16];
        Q += qbuf[(ww * 4 + nt) * 32 + lc] + qbuf[(ww * 4 + nt) * 32 + lc + 16];
      }
      statsPart[(size_t)blockIdx.x * 128 + t] = S;
      statsPart[(size_t)blockIdx.x * 128 + 64 + t] = Q;
    }
  }
}

// statOut[0..63]=mean, statOut[64..127]=rsqrt(var+eps). 64 threads.
__global__ void bn_finalize_kernel(const float* __restrict__ parts, int nblocks,
                                   int rowsValid, float* __restrict__ statOut) {
  int c = threadIdx.x;
  float S = 0.0f, Q = 0.0f;
  for (int i = 0; i < nblocks; ++i) {
    S += parts[(size_t)i * 128 + c];
    Q += parts[(size_t)i * 128 + 64 + c];
  }
  float inv = 1.0f / (float)rowsValid;
  float m = S * inv;
  float var = Q * inv - m * m;
  statOut[c] = m;
  statOut[64 + c] = rsqrtf(var + 1e-5f);
}

// ---------------------------------------------------------------------------
// Per-graph interaction. One 256-thread block per graph.
//   Hl = BN(yh rows), Sl = BN(ys rows), Pm = tanh(Hl @ Sl^T);
//   z[b] = [sum(h), colsum(Pm@Sl), sum(hs), colsum(Pm^T@Hl)] (256 floats).
// h2/hs2 never touch memory; only their segment sums are needed.
// ---------------------------------------------------------------------------
__global__ __launch_bounds__(256) void interact_kernel(
    const float* __restrict__ h_raw, const float* __restrict__ hs_raw,
    const float* __restrict__ yh, const float* __restrict__ ys,
    const float* __restrict__ statH, const float* __restrict__ statS,
    const float* __restrict__ gH, const float* __restrict__ bH,
    const float* __restrict__ gS, const float* __restrict__ bS,
    float* __restrict__ z) {
  __shared__ float Hl[96 * 65];
  __shared__ float Sl[64 * 65];
  __shared__ float Pm[96 * 65];
  __shared__ float ph2[12 * 64];
  __shared__ float ps2[8 * 64];
  __shared__ float sumH[4 * 64];
  __shared__ float sumS[4 * 64];

  int b = blockIdx.x;
  int t = threadIdx.x;
  // closed-form ragged offsets: mc = 32 + b%65, sc = 16 + b%49
  int qm = b / 65, rm = b - qm * 65;
  int moff = 32 * b + qm * 2080 + (rm * (rm - 1)) / 2;
  int mc = 32 + rm;
  int qs = b / 49, rs = b - qs * 49;
  int soff = 16 * b + qs * 1176 + (rs * (rs - 1)) / 2;
  int sc = 16 + rs;

  for (int idx = t; idx < 96 * 64; idx += 256) {
    int r = idx >> 6, c = idx & 63;
    float v = 0.0f;
    if (r < mc)
      v = (yh[(size_t)(moff + r) * HID + c] - statH[c]) * statH[64 + c] * gH[c] + bH[c];
    Hl[r * 65 + c] = v;
  }
  for (int idx = t; idx < 64 * 64; idx += 256) {
    int r = idx >> 6, c = idx & 63;
    float v = 0.0f;
    if (r < sc)
      v = (ys[(size_t)(soff + r) * HID + c] - statS[c]) * statS[64 + c] * gS[c] + bS[c];
    Sl[r * 65 + c] = v;
  }
  {  // raw-feature segment sums (4-way partials, fixed-order combine later)
    int c = t & 63, g = t >> 6;
    float aH = 0.0f, aS = 0.0f;
    for (int r = g; r < mc; r += 4) aH += h_raw[(size_t)(moff + r) * HID + c];
    for (int r = g; r < sc; r += 4) aS += hs_raw[(size_t)(soff + r) * HID + c];
    sumH[g * 64 + c] = aH;
    sumS[g * 64 + c] = aS;
  }
  __syncthreads();

  int w = t >> 5, lane = t & 31;
  int li = lane & 15, hi = lane >> 4, koff = hi * 2;

  // stage 1: Pm = tanh(Hl @ Sl^T), 6x4 tiles over 8 waves
  for (int tile = w; tile < 24; tile += 8) {
    int mt = tile >> 2, nt = tile & 3;
    v8f acc = v8f_zero();
#pragma unroll
    for (int k0 = 0; k0 < 64; k0 += 4) {
      v2f a, bb;
      a.x = Hl[(mt * 16 + li) * 65 + k0 + koff];
      a.y = Hl[(mt * 16 + li) * 65 + k0 + koff + 1];
      bb.x = Sl[(nt * 16 + li) * 65 + k0 + koff];
      bb.y = Sl[(nt * 16 + li) * 65 + k0 + koff + 1];
      acc = wmma_f32_k4(a, bb, acc);
    }
#pragma unroll
    for (int j = 0; j < 8; ++j)
      Pm[(mt * 16 + j + 8 * hi) * 65 + nt * 16 + li] = tanhf(acc[j]);
  }
  __syncthreads();

  // stage 2: h2 = Pm@Sl (tiles 0..23), hs2 = Pm^T@Hl (tiles 24..39)
  for (int tile = w; tile < 40; tile += 8) {
    if (tile < 24) {
      int mt = tile >> 2, nt = tile & 3;
      v8f acc = v8f_zero();
#pragma unroll
      for (int k0 = 0; k0 < 64; k0 += 4) {
        v2f a, bb;
        a.x = Pm[(mt * 16 + li) * 65 + k0 + koff];
        a.y = Pm[(mt * 16 + li) * 65 + k0 + koff + 1];
        bb.x = Sl[(k0 + koff) * 65 + nt * 16 + li];
        bb.y = Sl[(k0 + koff + 1) * 65 + nt * 16 + li];
        acc = wmma_f32_k4(a, bb, acc);
      }
      float p = 0.0f;
#pragma unroll
      for (int j = 0; j < 8; ++j) {
        int row = mt * 16 + j + 8 * hi;
        if (row < mc) p += acc[j];
      }
      ph2[(mt * 2 + hi) * 64 + nt * 16 + li] = p;
    } else {
      int s = tile - 24;
      int mt = s >> 2, nt = s & 3;
      v8f acc = v8f_zero();
#pragma unroll
      for (int k0 = 0; k0 < 96; k0 += 4) {
        v2f a, bb;
        a.x = Pm[(k0 + koff) * 65 + mt * 16 + li];  // Pm^T
        a.y = Pm[(k0 + koff + 1) * 65 + mt * 16 + li];
        bb.x = Hl[(k0 + koff) * 65 + nt * 16 + li];
        bb.y = Hl[(k0 + koff + 1) * 65 + nt * 16 + li];
        acc = wmma_f32_k4(a, bb, acc);
      }
      float p = 0.0f;
#pragma unroll
      for (int j = 0; j < 8; ++j) {
        int row = mt * 16 + j + 8 * hi;
        if (row < sc) p += acc[j];
      }
      ps2[(mt * 2 + hi) * 64 + nt * 16 + li] = p;
    }
  }
  __syncthreads();

  if (t < 64) {  // assemble z[b] with fixed-order sums
    float zh = sumH[t] + sumH[64 + t] + sumH[128 + t] + sumH[192 + t];
    float zs = sumS[t] + sumS[64 + t] + sumS[128 + t] + sumS[192 + t];
    float z2 = 0.0f;
    for (int i = 0; i < 12; ++i) z2 += ph2[i * 64 + t];
    float z3 = 0.0f;
    for (int i = 0; i < 8; ++i) z3 += ps2[i * 64 + t];
    float* zb = z + (size_t)b * 256;
    zb[t] = zh;
    zb[64 + t] = z2;
    zb[128 + t] = zs;
    zb[192 + t] = z3;
  }
}

// ---------------------------------------------------------------------------
// out1 = relu(Z @ l1_W + l1_b)   [4096,256] @ [256,256]
// Double-buffered LDS weight stream: async-prefetch chunk kc+1 while WMMAs
// consume chunk kc. End-of-iteration wait(ASYNCcnt)+barrier = safe overwrite.
// ---------------------------------------------------------------------------
__global__ __launch_bounds__(256) void mlp1_kernel(
    const float* __restrict__ Z, const float* __restrict__ W,
    const float* __restrict__ bias, float* __restrict__ out1) {
  __shared__ __align__(16) float Wl[2][64 * 256];  // 128 KB
  int t = threadIdx.x;
  int w = t >> 5, lane = t & 31;
  int li = lane & 15, hi = lane >> 4, koff = hi * 2;
  int mt = w & 1, ng = w >> 1;
  int r0 = blockIdx.x * 32 + mt * 16;

  for (int i = t; i < 4096; i += 256)  // chunk 0 (64x256 floats)
    stage_b128(&Wl[0][i * 4], &W[i * 4]);
  stage_commit();
  __syncthreads();

  v8f acc[4];
#pragma unroll
  for (int nt = 0; nt < 4; ++nt) acc[nt] = v8f_zero();

  for (int kc = 0; kc < 4; ++kc) {
    if (kc < 3) {  // prefetch next chunk into the other buffer
      const float* src = W + (size_t)(kc + 1) * 64 * 256;
      for (int i = t; i < 4096; i += 256)
        stage_b128(&Wl[(kc + 1) & 1][i * 4], &src[i * 4]);
    }
    const float* Wc = Wl[kc & 1];
    const float* Zr = Z + (size_t)(r0 + li) * 256 + kc * 64;
#pragma unroll
    for (int k0 = 0; k0 < 64; k0 += 4) {
      v2f a;
      a.x = Zr[k0 + koff];
      a.y = Zr[k0 + koff + 1];
#pragma unroll
      for (int nt = 0; nt < 4; ++nt) {
        int col = ng * 64 + nt * 16 + li;
        v2f bb;
        bb.x = Wc[(k0 + koff) * 256 + col];
        bb.y = Wc[(k0 + koff + 1) * 256 + col];
        acc[nt] = wmma_f32_k4(a, bb, acc[nt]);
      }
    }
    stage_commit();   // this wave's prefetch has landed
    __syncthreads();  // everyone done reading current buffer
  }
#pragma unroll
  for (int nt = 0; nt < 4; ++nt) {
    int col = ng * 64 + nt * 16 + li;
    float bv = bias[col];
#pragma unroll
    for (int j = 0; j < 8; ++j) {
      float v = fmaxf(acc[nt][j] + bv, 0.0f);
      out1[(size_t)(r0 + j + 8 * hi) * 256 + col] = v;
    }
  }
}

// out[g] = out1[g,:] . l2_W + l2_b
__global__ __launch_bounds__(256) void mlp2_kernel(
    const float* __restrict__ X, const float* __restrict__ w2,
    const float* __restrict__ b2, float* __restrict__ out) {
  __shared__ __align__(16) float wl[256];
  int t = threadIdx.x;
  if (t < 64) stage_b128(&wl[t * 4], &w2[t * 4]);
  stage_commit();
  __syncthreads();
  int g = blockIdx.x * 256 + t;
  const float* xr = X + (size_t)g * 256;
  float acc = b2[0];
  for (int c = 0; c < 256; ++c) acc += xr[c] * wl[c];
  out[g] = acc;
}

// ---------------------------------------------------------------------------
extern "C" void kernel_launch(void* const* d_in, const int* in_sizes, int n_in,
                              void* d_out, int out_size, void* d_ws, size_t ws_size,
                              hipStream_t stream) {
  (void)n_in; (void)out_size; (void)ws_size;
  const float* hidden = (const float*)d_in[0];
  const float* sfeats = (const float*)d_in[1];
  const float* embW = (const float*)d_in[2];
  const float* embB = (const float*)d_in[3];
  const float* gcnW = (const float*)d_in[4];
  const float* gcnB = (const float*)d_in[5];
  const float* hW = (const float*)d_in[6];
  const float* sW = (const float*)d_in[7];
  const float* bnHg = (const float*)d_in[8];
  const float* bnHb = (const float*)d_in[9];
  const float* bnSg = (const float*)d_in[10];
  const float* bnSb = (const float*)d_in[11];
  const float* l1W = (const float*)d_in[12];
  const float* l1b = (const float*)d_in[13];
  const float* l2W = (const float*)d_in[14];
  const float* l2b = (const float*)d_in[15];
  const int* sseg = (const int*)d_in[18];
  const int* spos = (const int*)d_in[19];
  float* out = (float*)d_out;

  int Nm = in_sizes[0] / HID;
  int Ns = in_sizes[1] / NODE_DIM;
  int NmP = (Nm + 127) & ~127;
  int NsP = (Ns + 127) & ~127;
  int nbH = NmP / 128, nbS = NsP / 128;
  int nbMax = nbH > nbS ? nbH : nbS;

  float* ws = (float*)d_ws;
  float* hsA = ws;                         // [NsP,64] GCN state
  float* hsB = hsA + (size_t)NsP * HID;    // [NsP,64] agg scratch, later ys
  float* yh = hsB + (size_t)NsP * HID;     // [NmP,64] hidden @ hW
  float* zbuf = yh + (size_t)NmP * HID;    // [4096,256]
  float* o1 = zbuf + (size_t)NGRAPH * 256; // [4096,256]
  float* parts = o1 + (size_t)NGRAPH * 256;
  float* statH = parts + (size_t)nbMax * 128;
  float* statS = statH + 128;

  // 1) embed
  embed_kernel<<<(Ns + 15) / 16, 256, 0, stream>>>(sfeats, embW, embB, hsA, Ns);
  // 2) GCN x4 (ring aggregate -> 64x64 WMMA matmul + bias + relu)
  for (int l = 0; l < 4; ++l) {
    agg_kernel<<<(Ns * HID + 255) / 256, 256, 0, stream>>>(hsA, sseg, spos, hsB, Ns);
    mm64_kernel<<<NsP / 128, 256, 0, stream>>>(hsB, gcnW + (size_t)l * HID * HID,
                                               gcnB + (size_t)l * HID, hsA,
                                               nullptr, Ns, 1);
  }
  // 3) yh = hidden @ hW  (+ BN stats), finalize
  mm64_kernel<<<NmP / 128, 256, 0, stream>>>(hidden, hW, nullptr, yh, parts, Nm, 0);
  bn_finalize_kernel<<<1, 64, 0, stream>>>(parts, nbH, Nm, statH);
  // 4) ys = hs @ sW  (+ BN stats), finalize   (ys reuses hsB)
  mm64_kernel<<<NsP / 128, 256, 0, stream>>>(hsA, sW, nullptr, hsB, parts, Ns, 0);
  bn_finalize_kernel<<<1, 64, 0, stream>>>(parts, nbS, Ns, statS);
  // 5) per-graph interaction -> z
  interact_kernel<<<NGRAPH, 256, 0, stream>>>(hidden, hsA, yh, hsB, statH, statS,
                                              bnHg, bnHb, bnSg, bnSb, zbuf);
  // 6) MLP
  mlp1_kernel<<<NGRAPH / 32, 256, 0, stream>>>(zbuf, l1W, l1b, o1);
  mlp2_kernel<<<NGRAPH / 256, 256, 0, stream>>>(o1, l2W, l2b, out);
}